// KuramotoLayer_8229157339891
// MI455X (gfx1250) — compile-verified
//
#include <hip/hip_runtime.h>
#include <math.h>

// Kuramoto coupling on MI455X (gfx1250), factored into WMMA f32 GEMMs:
//   sin(tj - ti - a) = sin tj * cos(ti+a) - cos tj * sin(ti+a)
//   coupling[i,d] = cos ti_d * (P.S - Q.C) - sin ti_d * (Q.S + P.C)
//   P = W*cos(alpha), Q = W*sin(alpha); S/C = sin/cos of theta_j per dim.
// Memory-bound: 268 MB of W/alpha streamed once (~11.5 us @ 23.3 TB/s).

#define NB 2
#define NN 4096
#define ND 4
#define NGROUP (NN / 4)        // 1024 K-groups of 4 j's per batch row
#define TILES_PER_B (NN / 16)  // 256 16-row output tiles per batch

typedef float v2f __attribute__((ext_vector_type(2)));
typedef float v8f __attribute__((ext_vector_type(8)));

// ---------------------------------------------------------------------------
// Kernel 1: build B-operand panel in V_WMMA_F32_16X16X4_F32 lane layout.
// B is K(4) x N(16): lane l holds (k = 2*(l>>4)+r, n = l&15) in VGPR r.
// Columns: n<4 -> sin(theta[j][n]); 4<=n<8 -> cos(theta[j][n-4]); else 0.
// One float2 per (batch, group, lane): 2*1024*32*8B = 512 KB in d_ws.
// ---------------------------------------------------------------------------
__global__ void build_bpanel(const float* __restrict__ theta,
                             float2* __restrict__ bpanel) {
    int tid  = blockIdx.x * blockDim.x + threadIdx.x;  // 65536 threads total
    int lane = tid & 31;
    int g    = (tid >> 5) & (NGROUP - 1);
    int b    = tid >> 15;
    int n    = lane & 15;
    int kh   = lane >> 4;
    int j0   = 4 * g + 2 * kh;

    float2 v;
    if (n < 4) {
        v.x = __sinf(theta[(b * NN + j0)     * ND + n]);
        v.y = __sinf(theta[(b * NN + j0 + 1) * ND + n]);
    } else if (n < 8) {
        v.x = __cosf(theta[(b * NN + j0)     * ND + (n - 4)]);
        v.y = __cosf(theta[(b * NN + j0 + 1) * ND + (n - 4)]);
    } else {
        v.x = 0.0f;
        v.y = 0.0f;
    }
    bpanel[tid] = v;  // index == ((b*NGROUP + g)*32 + lane)
}

// ---------------------------------------------------------------------------
// Kernel 2: main. 512 blocks (batch x 16-row tile), 8 wave32 per block.
// Wave w owns K-slice j in [w*512, w*512+512): 128 iterations of K=4.
// Per iteration/lane: float2 of W, alpha (A-operand layout), float2 B-panel,
// sincos(alpha) -> P,Q operands -> two chained v_wmma_f32_16x16x4_f32.
// Unconditional speculative stream-prefetch (single-basic-block inner loop).
// Cross-wave LDS reduction, then 16 lanes finalize + L2-normalize.
// ---------------------------------------------------------------------------
__global__ void __launch_bounds__(256) kuramoto_main(
    const float* __restrict__ theta, const float* __restrict__ gamma,
    const float* __restrict__ W, const float* __restrict__ alpha,
    const float2* __restrict__ bpanel, float* __restrict__ out) {

    __shared__ float red[8][2][256];  // 8 waves x {P,Q} x 16x16 tile

    const int b     = blockIdx.x >> 8;               // 256 tiles per batch
    const int itile = blockIdx.x & (TILES_PER_B - 1);
    const int lane  = threadIdx.x & 31;
    const int wv    = threadIdx.x >> 5;              // wave 0..7
    const int m     = lane & 15;                     // A row / B col
    const int kh    = lane >> 4;                     // K half-select
    const int row   = itile * 16 + m;

    const size_t matbase = (size_t)b * NN * NN + (size_t)row * NN;
    // float2 index within row: j/2; this wave starts at j = wv*512, +2*kh.
    const float2* wp = (const float2*)(W + matbase)     + (size_t)wv * 256 + kh;
    const float2* ap = (const float2*)(alpha + matbase) + (size_t)wv * 256 + kh;
    const float2* bp = bpanel + ((size_t)b * NGROUP + (size_t)wv * 128) * 32 + lane;

    v8f accP = {0.f, 0.f, 0.f, 0.f, 0.f, 0.f, 0.f, 0.f};
    v8f accQ = {0.f, 0.f, 0.f, 0.f, 0.f, 0.f, 0.f, 0.f};

    for (int g = 0; g < 128; ++g) {
        float2 w2 = wp[2 * g];
        float2 a2 = ap[2 * g];
        float2 bv = bp[g * 32];

        // Speculative stream-prefetch 16 iterations (256 B/lane) ahead.
        // global_prefetch_b8 at DEV scope is speculative: a tail prefetch
        // past the allocation is silently dropped, so no bounds guard ->
        // keeps the inner loop a single basic block.
        __builtin_prefetch(&wp[2 * (g + 16)], 0, 1);
        __builtin_prefetch(&ap[2 * (g + 16)], 0, 1);

        float ca0 = __cosf(a2.x), ca1 = __cosf(a2.y);
        float sa0 = __sinf(a2.x), sa1 = __sinf(a2.y);

        v2f pa, qa, bb;
        pa.x = w2.x * ca0;  pa.y = w2.y * ca1;   // P = W*cos(alpha)
        qa.x = w2.x * sa0;  qa.y = w2.y * sa1;   // Q = W*sin(alpha)
        bb.x = bv.x;        bb.y = bv.y;

        accP = __builtin_amdgcn_wmma_f32_16x16x4_f32(
            false, pa, false, bb, (short)0, accP, false, false);
        accQ = __builtin_amdgcn_wmma_f32_16x16x4_f32(
            false, qa, false, bb, (short)0, accQ, false, false);
    }

    // C/D layout: lane l, VGPR r -> element (m = r + 8*(l>>4), n = l&15).
#pragma unroll
    for (int r = 0; r < 8; ++r) {
        red[wv][0][(r + 8 * kh) * 16 + m] = accP[r];
        red[wv][1][(r + 8 * kh) * 16 + m] = accQ[r];
    }
    __syncthreads();

    // Cross-wave sum: each thread owns exactly one tile element -> no races.
    {
        int e = threadIdx.x;
        float sp = 0.f, sq = 0.f;
#pragma unroll
        for (int ww = 0; ww < 8; ++ww) {
            sp += red[ww][0][e];
            sq += red[ww][1][e];
        }
        red[0][0][e] = sp;
        red[0][1][e] = sq;
    }
    __syncthreads();

    // Finalize: one lane per output row.
    if (threadIdx.x < 16) {
        int mrow = threadIdx.x;
        int i    = itile * 16 + mrow;
        float4 th = ((const float4*)theta)[b * NN + i];
        float4 gm = ((const float4*)gamma)[b * NN + i];
        float thv[4] = {th.x, th.y, th.z, th.w};
        float gv[4]  = {gm.x, gm.y, gm.z, gm.w};
        float tn[4];
        float ss = 0.f;
#pragma unroll
        for (int d = 0; d < 4; ++d) {
            float s  = __sinf(thv[d]);
            float c  = __cosf(thv[d]);
            float PS = red[0][0][mrow * 16 + d];
            float PC = red[0][0][mrow * 16 + d + 4];
            float QS = red[0][1][mrow * 16 + d];
            float QC = red[0][1][mrow * 16 + d + 4];
            float coup = c * (PS - QC) - s * (QS + PC);
            float v = gv[d] + coup * (1.0f / (float)NN);  // theta+(gamma-theta)+coupling
            tn[d] = v;
            ss += v * v;
        }
        float inv = 1.0f / fmaxf(sqrtf(ss), 1e-6f);
        float4 o = {tn[0] * inv, tn[1] * inv, tn[2] * inv, tn[3] * inv};
        ((float4*)out)[b * NN + i] = o;
    }
}

extern "C" void kernel_launch(void* const* d_in, const int* in_sizes, int n_in,
                              void* d_out, int out_size, void* d_ws, size_t ws_size,
                              hipStream_t stream) {
    (void)in_sizes; (void)n_in; (void)out_size; (void)ws_size;
    const float* theta = (const float*)d_in[0];
    const float* gamma = (const float*)d_in[1];
    const float* W     = (const float*)d_in[2];
    const float* alpha = (const float*)d_in[3];
    float2* bpanel = (float2*)d_ws;  // 512 KB scratch, rebuilt every call

    build_bpanel<<<(NB * NGROUP * 32) / 256, 256, 0, stream>>>(theta, bpanel);
    kuramoto_main<<<NB * TILES_PER_B, 256, 0, stream>>>(
        theta, gamma, W, alpha, bpanel, (float*)d_out);
}